// GRUSeq2seqForecaster_44513041056182
// MI455X (gfx1250) — compile-verified
//
#include <hip/hip_runtime.h>
#include <cmath>

typedef __attribute__((ext_vector_type(2))) float v2f;
typedef __attribute__((ext_vector_type(8))) float v8f;

#define NB    4096
#define T_IN  168
#define NC    32
#define NH    64
#define NG    192
#define T_OUT 24
#define RS    68            // LDS row stride (floats) for h/prev tiles: conflict-free
#define NWAVES 4
#define NTHREADS (NWAVES * 32)

// ---- LDS layout (float offsets) ----
#define ENC_WIH 0                          // 12 n-tiles * 8 k-tiles * 64 = 6144
#define ENC_WHH (ENC_WIH + 12*8*64)        // 12 * 16 * 64 = 12288
#define DEC_WIH (ENC_WHH + 12*16*64)
#define DEC_WHH (DEC_WIH + 12*8*64)
#define PROJW   (DEC_WHH + 12*16*64)       // 2 * 16 * 64 = 2048
#define ENC_BRZ (PROJW + 2*16*64)          // 128 (bih+bhh, r|z cols)
#define ENC_BIN (ENC_BRZ + 128)            // 64
#define ENC_BHN (ENC_BIN + 64)             // 64
#define DEC_BRZ (ENC_BHN + 64)
#define DEC_BIN (DEC_BRZ + 128)
#define DEC_BHN (DEC_BIN + 64)
#define PROJ_B  (DEC_BHN + 64)             // 32
#define H_OFF   (PROJ_B + 32)
#define PV_OFF  (H_OFF + NWAVES*16*RS)
#define SMEM_FLOATS (PV_OFF + NWAVES*16*RS)
#define SMEM_BYTES  (SMEM_FLOATS * 4)      // = 192640 bytes

// fp32 WMMA: D(16x16,f32) = A(16x4,f32) * B(4x16,f32) + C
__device__ __forceinline__ v8f wmma4(v2f a, v2f b, v8f c) {
    return __builtin_amdgcn_wmma_f32_16x16x4_f32(false, a, false, b,
                                                 (short)0, c, false, false);
}

// Scatter W^T (Kdim x Ncols) into LDS in B-fragment order.
// slot = n_tile*k_tiles + kt; within slot: lane*2 + v.
// B-frag layout assumption (mirror of documented A layout):
//   lanes 0-15: N=lane, K = kt*4 + v ; lanes 16-31: N=lane-16, K = kt*4 + 2 + v
// Source W is row-major (Ncols rows x ld cols): element = W[N*ld + K].
__device__ __forceinline__ void init_frags(float* dst, const float* W,
                                           int n_tiles, int k_tiles, int ld,
                                           int tid) {
    int total = n_tiles * k_tiles * 64;
    for (int idx = tid; idx < total; idx += NTHREADS) {
        int slot = idx >> 6;
        int pos  = idx & 63;
        int ln   = pos >> 1;
        int v    = pos & 1;
        int nt   = slot / k_tiles;
        int kt   = slot - nt * k_tiles;
        int N    = nt * 16 + (ln & 15);
        int K    = kt * 4 + ((ln & 16) ? 2 : 0) + v;
        dst[idx] = W[N * ld + K];
    }
}

// One GRU step for a 16-row batch tile resident in LDS (hb, 16 x RS).
// xA: 8 A-fragments of the 16x32 input tile.
__device__ __forceinline__ void gru_step(const v2f* xA,
                                         const float* Wih, const float* Whh,
                                         const float* brz, const float* bin,
                                         const float* bhn,
                                         float* hb, int lane) {
    const int mrow  = lane & 15;
    const int koff  = (lane & 16) ? 2 : 0;
    const int lane2 = lane * 2;

    // h_{t-1} as A-fragments (16x64 -> 16 k-tiles)
    v2f hA[16];
#pragma unroll
    for (int kt = 0; kt < 16; ++kt)
        hA[kt] = *(const v2f*)&hb[mrow * RS + kt * 4 + koff];
    __builtin_amdgcn_wave_barrier();

#pragma unroll
    for (int j = 0; j < 4; ++j) {           // 4 column tiles per gate (H=64)
        float br = brz[j * 16 + mrow];
        float bz = brz[64 + j * 16 + mrow];
        float bi = bin[j * 16 + mrow];
        float bh = bhn[j * 16 + mrow];
        v8f aR = {br, br, br, br, br, br, br, br};
        v8f aZ = {bz, bz, bz, bz, bz, bz, bz, bz};
        v8f aI = {bi, bi, bi, bi, bi, bi, bi, bi};
        v8f aH = {bh, bh, bh, bh, bh, bh, bh, bh};

        // gi = x * Wih^T  (K = 32)
#pragma unroll
        for (int kt = 0; kt < 8; ++kt) {
            v2f bR = *(const v2f*)&Wih[((j    ) * 8 + kt) * 64 + lane2];
            v2f bZ = *(const v2f*)&Wih[((4 + j) * 8 + kt) * 64 + lane2];
            v2f bI = *(const v2f*)&Wih[((8 + j) * 8 + kt) * 64 + lane2];
            aR = wmma4(xA[kt], bR, aR);
            aZ = wmma4(xA[kt], bZ, aZ);
            aI = wmma4(xA[kt], bI, aI);
        }
        // gh = h * Whh^T  (K = 64); r/z accumulate on top, hn separate
#pragma unroll
        for (int kt = 0; kt < 16; ++kt) {
            v2f bR = *(const v2f*)&Whh[((j    ) * 16 + kt) * 64 + lane2];
            v2f bZ = *(const v2f*)&Whh[((4 + j) * 16 + kt) * 64 + lane2];
            v2f bH = *(const v2f*)&Whh[((8 + j) * 16 + kt) * 64 + lane2];
            aR = wmma4(hA[kt], bR, aR);
            aZ = wmma4(hA[kt], bZ, aZ);
            aH = wmma4(hA[kt], bH, aH);
        }

        // gate math in D layout: lane holds rows M = v + (lane<16?0:8), col j*16+mrow
        int mb = (lane & 16) ? 8 : 0;
#pragma unroll
        for (int v = 0; v < 8; ++v) {
            float hold = hb[(mb + v) * RS + j * 16 + mrow];
            float r = 1.0f / (1.0f + __expf(-aR[v]));
            float z = 1.0f / (1.0f + __expf(-aZ[v]));
            float n = tanhf(aI[v] + r * aH[v]);
            hb[(mb + v) * RS + j * 16 + mrow] = (1.0f - z) * n + z * hold;
        }
    }
    __builtin_amdgcn_wave_barrier();
}

__global__ __launch_bounds__(NTHREADS, 1)
void gru_forecast_kernel(const float* __restrict__ x,
                         const float* __restrict__ eWih, const float* __restrict__ eWhh,
                         const float* __restrict__ ebih, const float* __restrict__ ebhh,
                         const float* __restrict__ dWih, const float* __restrict__ dWhh,
                         const float* __restrict__ dbih, const float* __restrict__ dbhh,
                         const float* __restrict__ pW,   const float* __restrict__ pb,
                         float* __restrict__ out) {
    extern __shared__ float sm[];
    const int tid = threadIdx.x;

    // ---- stage weights into fragment-ordered LDS (once per WG) ----
    init_frags(sm + ENC_WIH, eWih, 12,  8, NC, tid);
    init_frags(sm + ENC_WHH, eWhh, 12, 16, NH, tid);
    init_frags(sm + DEC_WIH, dWih, 12,  8, NC, tid);
    init_frags(sm + DEC_WHH, dWhh, 12, 16, NH, tid);
    init_frags(sm + PROJW,   pW,    2, 16, NH, tid);
    for (int c = tid; c < 128; c += NTHREADS) {
        sm[ENC_BRZ + c] = ebih[c] + ebhh[c];
        sm[DEC_BRZ + c] = dbih[c] + dbhh[c];
    }
    for (int c = tid; c < 64; c += NTHREADS) {
        sm[ENC_BIN + c] = ebih[128 + c];
        sm[ENC_BHN + c] = ebhh[128 + c];
        sm[DEC_BIN + c] = dbih[128 + c];
        sm[DEC_BHN + c] = dbhh[128 + c];
    }
    for (int c = tid; c < 32; c += NTHREADS) sm[PROJ_B + c] = pb[c];

    const int lane = tid & 31;
    const int wave = tid >> 5;
    const int row0 = blockIdx.x * (NWAVES * 16) + wave * 16;
    float* hb = sm + H_OFF  + wave * 16 * RS;
    float* pv = sm + PV_OFF + wave * 16 * RS;
    const int mrow = lane & 15;
    const int koff = (lane & 16) ? 2 : 0;
    const int lane2 = lane * 2;

    // h0 = 0 ; prev = x[:, T_IN-1, :]
    for (int i = lane; i < 16 * RS; i += 32) hb[i] = 0.0f;
    for (int i = lane; i < 16 * NC; i += 32) {
        int r = i >> 5;
        int c = i & 31;
        pv[r * RS + c] = x[(size_t)(row0 + r) * (T_IN * NC) + (size_t)(T_IN - 1) * NC + c];
    }
    __syncthreads();

    // ---- encoder: 168 recurrent steps ----
    const float* xrow = x + (size_t)(row0 + mrow) * (T_IN * NC);
#pragma unroll 1
    for (int t = 0; t < T_IN; ++t) {
        v2f xA[8];
#pragma unroll
        for (int kt = 0; kt < 8; ++kt)
            xA[kt] = *(const v2f*)&xrow[t * NC + kt * 4 + koff];
        gru_step(xA, sm + ENC_WIH, sm + ENC_WHH,
                 sm + ENC_BRZ, sm + ENC_BIN, sm + ENC_BHN, hb, lane);
    }

    // ---- decoder: 24 autoregressive steps + projection ----
#pragma unroll 1
    for (int t = 0; t < T_OUT; ++t) {
        v2f pA[8];
#pragma unroll
        for (int kt = 0; kt < 8; ++kt)
            pA[kt] = *(const v2f*)&pv[mrow * RS + kt * 4 + koff];
        __builtin_amdgcn_wave_barrier();
        gru_step(pA, sm + DEC_WIH, sm + DEC_WHH,
                 sm + DEC_BRZ, sm + DEC_BIN, sm + DEC_BHN, hb, lane);

        // pred = h * projW^T + pb   (16x64 · 64x32)
        v2f hA[16];
#pragma unroll
        for (int kt = 0; kt < 16; ++kt)
            hA[kt] = *(const v2f*)&hb[mrow * RS + kt * 4 + koff];
        __builtin_amdgcn_wave_barrier();
#pragma unroll
        for (int j = 0; j < 2; ++j) {
            float b0 = sm[PROJ_B + j * 16 + mrow];
            v8f acc = {b0, b0, b0, b0, b0, b0, b0, b0};
#pragma unroll
            for (int kt = 0; kt < 16; ++kt) {
                v2f bP = *(const v2f*)&sm[PROJW + (j * 16 + kt) * 64 + lane2];
                acc = wmma4(hA[kt], bP, acc);
            }
            int mb = (lane & 16) ? 8 : 0;
#pragma unroll
            for (int v = 0; v < 8; ++v) {
                float p = acc[v];
                out[(size_t)(row0 + mb + v) * (T_OUT * NC) + (size_t)t * NC + j * 16 + mrow] = p;
                pv[(mb + v) * RS + j * 16 + mrow] = p;   // feed back as next input
            }
        }
        __builtin_amdgcn_wave_barrier();
    }
}

extern "C" void kernel_launch(void* const* d_in, const int* in_sizes, int n_in,
                              void* d_out, int out_size, void* d_ws, size_t ws_size,
                              hipStream_t stream) {
    const float* x    = (const float*)d_in[0];
    const float* eWih = (const float*)d_in[1];
    const float* eWhh = (const float*)d_in[2];
    const float* ebih = (const float*)d_in[3];
    const float* ebhh = (const float*)d_in[4];
    const float* dWih = (const float*)d_in[5];
    const float* dWhh = (const float*)d_in[6];
    const float* dbih = (const float*)d_in[7];
    const float* dbhh = (const float*)d_in[8];
    const float* pW   = (const float*)d_in[9];
    const float* pb   = (const float*)d_in[10];
    float* out = (float*)d_out;

    (void)in_sizes; (void)n_in; (void)out_size; (void)d_ws; (void)ws_size;

    hipFuncSetAttribute((const void*)gru_forecast_kernel,
                        hipFuncAttributeMaxDynamicSharedMemorySize, SMEM_BYTES);

    dim3 grid(NB / (NWAVES * 16));   // 64 workgroups
    dim3 block(NTHREADS);            // 128 threads = 4 waves
    gru_forecast_kernel<<<grid, block, SMEM_BYTES, stream>>>(
        x, eWih, eWhh, ebih, ebhh, dWih, dWhh, dbih, dbhh, pW, pb, out);
}